// PackedINTAsymmetricWeightsDequantizer_29274497089864
// MI455X (gfx1250) — compile-verified
//
#include <hip/hip_runtime.h>

// PackedINTAsymmetricWeightsDequantizer for MI455X (gfx1250).
//
// Roofline: 128 MB in + 256 MB out @ 23.3 TB/s => ~16.5 us floor; pure
// streaming, write-dominated. No matrix math => no WMMA. We use the CDNA5
// async tensor-class path (GLOBAL_LOAD_ASYNC_TO_LDS_B128 + ASYNCcnt) to
// double-buffer the packed weights through LDS, and 128-bit non-temporal
// stores for the 256 MB streamed output (working set > 192 MB L2).

#define OUT_F        8192
#define PACKED_ROW   4096                     // int32 per row (1 byte/elem)
#define TOTAL_PACKED (OUT_F * PACKED_ROW)     // 33,554,432 int32
#define THREADS      256
#define TILE_INTS    1024                     // 256 threads x 4 int32 (b128)
#define NTILES       (TOTAL_PACKED / TILE_INTS)   // 32768
#define NBLOCKS      4096                     // 8 tiles per block (grid-stride)

typedef __attribute__((ext_vector_type(4))) float v4f;
typedef __attribute__((ext_vector_type(4))) int   v4i;

__global__ __launch_bounds__(THREADS)
void dequant_u4_asym_kernel(const int* __restrict__ weight,
                            const float* __restrict__ scale,
                            const float* __restrict__ zero_point,
                            float* __restrict__ out) {
    // Ping-pong staging buffers: each lane owns a private 16B slot, written
    // by the async engine and read back by the same lane (no barriers).
    __shared__ int lds[2][TILE_INTS];

    const int tid = threadIdx.x;
    // Low 32 bits of a generic pointer into LDS are the LDS byte address
    // (aperture lives in the high 32 bits) -> usable as the async VDST.
    const unsigned lds_a0 = (unsigned)(uintptr_t)(&lds[0][tid * 4]);
    const unsigned lds_a1 = (unsigned)(uintptr_t)(&lds[1][tid * 4]);

    int tile = (int)blockIdx.x;

    // Prologue: async-stage the first tile into buffer 0.
    {
        const int* g = weight + (size_t)tile * TILE_INTS + tid * 4;
        asm volatile("global_load_async_to_lds_b128 %0, %1, off"
                     :: "v"(lds_a0), "v"(g) : "memory");
    }

    int buf = 0;
    for (; tile < NTILES; tile += NBLOCKS) {
        const int next = tile + NBLOCKS;
        if (next < NTILES) {
            // Prefetch the next tile into the other buffer, then wait for
            // the current tile only (async loads complete in order).
            const int* g = weight + (size_t)next * TILE_INTS + tid * 4;
            const unsigned lb = buf ? lds_a0 : lds_a1;
            asm volatile("global_load_async_to_lds_b128 %0, %1, off"
                         :: "v"(lb), "v"(g) : "memory");
            asm volatile("s_wait_asynccnt 0x1" ::: "memory");
        } else {
            asm volatile("s_wait_asynccnt 0x0" ::: "memory");
        }

        // Each 1024-int32 tile lies inside one output row (4096 packed/row),
        // so row is block-uniform -> scalar loads of scale / zero_point.
        const int   row = tile >> 2;
        const float sv  = scale[row];
        const float zv  = zero_point[row];

        // Read back this lane's 16 bytes of packed data (ds_load_b128).
        const v4i w = *(const v4i*)(&lds[buf][tid * 4]);

        // Unpack: low nibble -> even output, high nibble -> odd output.
        // Exact reference numerics: (float(q) - zp) * s  (sub then mul).
        v4f o0, o1;
        o0.x = ((float)( w.x       & 0xF) - zv) * sv;
        o0.y = ((float)((w.x >> 4) & 0xF) - zv) * sv;
        o0.z = ((float)( w.y       & 0xF) - zv) * sv;
        o0.w = ((float)((w.y >> 4) & 0xF) - zv) * sv;
        o1.x = ((float)( w.z       & 0xF) - zv) * sv;
        o1.y = ((float)((w.z >> 4) & 0xF) - zv) * sv;
        o1.z = ((float)( w.w       & 0xF) - zv) * sv;
        o1.w = ((float)((w.w >> 4) & 0xF) - zv) * sv;

        // Packed flat index p -> output flat 2p / 2p+1: this lane's 8 floats
        // are contiguous. Non-temporal 128-bit stores (streamed 256 MB).
        v4f* op = (v4f*)(out + (size_t)tile * (TILE_INTS * 2) + tid * 8);
        __builtin_nontemporal_store(o0, op);
        __builtin_nontemporal_store(o1, op + 1);

        buf ^= 1;
    }
}

extern "C" void kernel_launch(void* const* d_in, const int* in_sizes, int n_in,
                              void* d_out, int out_size, void* d_ws, size_t ws_size,
                              hipStream_t stream) {
    const int*   weight     = (const int*)d_in[0];     // [8192, 4096] int32
    const float* scale      = (const float*)d_in[1];   // [8192, 1] f32
    const float* zero_point = (const float*)d_in[2];   // [8192, 1] f32
    float*       out        = (float*)d_out;           // [8192, 8192] f32

    (void)in_sizes; (void)n_in; (void)out_size; (void)d_ws; (void)ws_size;

    dequant_u4_asym_kernel<<<NBLOCKS, THREADS, 0, stream>>>(
        weight, scale, zero_point, out);
}